// KVLora_27539330302341
// MI455X (gfx1250) — compile-verified
//
#include <hip/hip_runtime.h>

typedef __attribute__((ext_vector_type(2))) float v2f;
typedef __attribute__((ext_vector_type(8))) float v8f;

// Problem dimensions (match reference: L,B,H,T,D,R = 16,1,32,1024,128,16)
constexpr int L = 16;
constexpr int H = 32;
constexpr int T = 1024;
constexpr int D = 128;
constexpr int R = 16;

constexpr int TCHUNK = 128;          // t-rows handled per workgroup (8 waves x 16)
constexpr int NTCH = T / TCHUNK;     // 8 chunks along T

// One wave computes a 16x16 (t x d) tile:
//   delta_tile = A_tile^T (16x16 over r) applied as 4 chained K=4 f32 WMMAs,
// then streams kv tile through registers: out = kv + delta.
__global__ __launch_bounds__(256) void kvlora_wmma_f32(
    const float* __restrict__ keys,
    const float* __restrict__ vals,
    const float* __restrict__ key_A,
    const float* __restrict__ key_B,
    const float* __restrict__ value_A,
    const float* __restrict__ value_B,
    float* __restrict__ out)
{
    int bid = blockIdx.x;
    const int tch = bid % NTCH;  bid /= NTCH;
    const int h   = bid % H;     bid /= H;
    const int l   = bid % L;
    const int s   = bid / L;     // 0 = keys, 1 = vals

    const float* __restrict__ A  = (s == 0 ? key_A : value_A) + (size_t)l * R * T;
    const float* __restrict__ Bm = (s == 0 ? key_B : value_B)
                                   + ((size_t)l * H + h) * (size_t)D * R;
    const float* __restrict__ kv = (s == 0 ? keys : vals)
                                   + ((size_t)l * H + h) * (size_t)T * D;
    float* __restrict__ outp = out + ((((size_t)s * L + l) * H + h) * (size_t)T) * D;

    const int lane = threadIdx.x & 31;
    const int wave = threadIdx.x >> 5;
    const int t0   = tch * TCHUNK + wave * 16;

    const int m    = lane & 15;   // row (A: M=t) / col (B: N=d) within tile
    const int hi   = lane >> 4;   // 0 or 1 (half-wave)
    const int roff = hi * 2;      // K sub-offset per 16x4 f32 operand layout

    // ---- A fragments: 4 sub-blocks of K=4 covering r = 0..15 (reused for all d-tiles)
    // ISA 16x4 f32 A layout: lanes 0-15 M=0..15 {VGPR0:K=0, VGPR1:K=1},
    //                        lanes16-31 M=0..15 {VGPR0:K=2, VGPR1:K=3}
    v2f afrag[4];
#pragma unroll
    for (int k0 = 0; k0 < 4; ++k0) {
        const float* ap = A + (size_t)(4 * k0 + roff) * T + (t0 + m);
        afrag[k0].x = ap[0];
        afrag[k0].y = ap[T];
    }

#pragma unroll
    for (int dt = 0; dt < D / 16; ++dt) {
        const int d0 = dt * 16;

        // ---- B fragments (K=r, N=d); r is innermost in memory (contiguous pair loads)
        const float* bp = Bm + (size_t)(d0 + m) * R;
        v8f acc = {};
#pragma unroll
        for (int k0 = 0; k0 < 4; ++k0) {
            v2f bfrag;
            bfrag.x = bp[4 * k0 + roff];
            bfrag.y = bp[4 * k0 + roff + 1];
            // D = A x B + C ; chained accumulation over R=16 in 4 K=4 steps
            acc = __builtin_amdgcn_wmma_f32_16x16x4_f32(
                /*neg_a=*/false, afrag[k0],
                /*neg_b=*/false, bfrag,
                /*c_mod=*/(short)0, acc,
                /*reuse_a=*/false, /*reuse_b=*/false);
        }

        // ---- Stream kv tile through: out = kv + delta (non-temporal both ways:
        // each byte touched exactly once; 537MB stream >> useful L2 residency)
        // C/D f32 layout: VGPR i -> row t0 + i + 8*hi, col d0 + (lane&15)
        const size_t base = (size_t)(t0 + hi * 8) * D + (d0 + m);
#pragma unroll
        for (int i = 0; i < 8; ++i) {
            const size_t idx = base + (size_t)i * D;
            float v = __builtin_nontemporal_load(kv + idx);
            __builtin_nontemporal_store(v + acc[i], outp + idx);
        }
    }
}

extern "C" void kernel_launch(void* const* d_in, const int* in_sizes, int n_in,
                              void* d_out, int out_size, void* d_ws, size_t ws_size,
                              hipStream_t stream) {
    (void)in_sizes; (void)n_in; (void)out_size; (void)d_ws; (void)ws_size;
    const float* keys    = (const float*)d_in[0];
    const float* vals    = (const float*)d_in[1];
    const float* key_A   = (const float*)d_in[2];
    const float* key_B   = (const float*)d_in[3];
    const float* value_A = (const float*)d_in[4];
    const float* value_B = (const float*)d_in[5];

    const int nblocks = 2 * L * H * NTCH;  // 8192 workgroups, 256 threads (8 waves)
    kvlora_wmma_f32<<<nblocks, 256, 0, stream>>>(
        keys, vals, key_A, key_B, value_A, value_B, (float*)d_out);
}